// S3RecModel_31001073943401
// MI455X (gfx1250) — compile-verified
//
#include <hip/hip_runtime.h>

// Problem constants (from reference)
#define B_ 64
#define L_ 2048
#define D_ 256
#define C_ 256

// Native clang vector type: accepted by __builtin_nontemporal_* and gives
// b128 vector memory ops. Supports .x/.y/.z/.w swizzles.
typedef float v4f __attribute__((ext_vector_type(4)));

constexpr int kBlock       = 1024;              // 32 waves (wave32)
constexpr int kRowsPerIter = 32;                // one wave per row
constexpr int kIters       = L_ / kRowsPerIter; // 64

// Dynamic-LDS layout in float units (blob starts at LDS byte offset 0:
// no static __shared__ in this TU, so the extern shared array begins at 0,
// which the async-to-LDS byte offsets below rely on).
constexpr int POS_F  = 0;                           // 32 rows * 256 = 8192 floats (32 KB)
constexpr int SUM_F  = POS_F + kRowsPerIter * D_;   // 8192
constexpr int CNT_F  = SUM_F + C_ * D_;             // 73728
constexpr int SMEM_F = CNT_F + C_;                  // 73984 floats
constexpr size_t SMEM_BYTES = (size_t)SMEM_F * 4;   // 295,936 B <= 320 KB

extern "C" __global__ __launch_bounds__(kBlock, 1)
void s3rec_fused(const int* __restrict__ ids,
                 const int* __restrict__ labels,
                 const float* __restrict__ rq,
                 const float* __restrict__ itab,
                 const float* __restrict__ ptab,
                 const float* __restrict__ gamma,
                 const float* __restrict__ beta,
                 float* __restrict__ out)
{
    extern __shared__ float smem[];

    const int b    = blockIdx.x;        // one workgroup per batch
    const int tid  = threadIdx.x;
    const int wid  = tid >> 5;          // wave id 0..31  -> row within iter
    const int lane = tid & 31;          // lane 0..31     -> 8 floats of the row
    const int col  = lane * 8;

    // Zero the cluster accumulator + counts
    for (int i = SUM_F + tid; i < SMEM_F; i += kBlock) smem[i] = 0.0f;

    // gamma/beta are reused every row: keep this lane's 8 values in registers
    const v4f g0 = *(const v4f*)(gamma + col);
    const v4f g1 = *(const v4f*)(gamma + col + 4);
    const v4f e0 = *(const v4f*)(beta  + col);
    const v4f e1 = *(const v4f*)(beta  + col + 4);

    for (int it = 0; it < kIters; ++it) {
        // ---- async-stage 32 contiguous pos_table rows (32 KB) into LDS ----
        {
            const unsigned long long src =
                (unsigned long long)(const void*)(ptab + (size_t)it * kRowsPerIter * D_);
#pragma unroll
            for (int k = 0; k < 2; ++k) {
                const unsigned chunk  = (unsigned)tid + (unsigned)k * kBlock; // 0..2047 x 16B
                const unsigned ldsoff = (unsigned)(POS_F * 4) + chunk * 16u;
                const unsigned long long ga = src + (unsigned long long)chunk * 16ull;
                asm volatile("global_load_async_to_lds_b128 %0, %1, off"
                             :: "v"(ldsoff), "v"(ga) : "memory");
            }
            asm volatile("s_wait_asynccnt 0" ::: "memory");
        }
        __syncthreads();   // staged pos visible to all waves; zeroing also covered (it==0)

        const int    l      = it * kRowsPerIter + wid;
        const size_t rowidx = (size_t)b * L_ + l;

        // streaming read-once input: nontemporal (b128 NT loads)
        const float* rqp = rq + rowidx * D_ + col;
        const v4f r0 = __builtin_nontemporal_load((const v4f*)rqp);
        const v4f r1 = __builtin_nontemporal_load((const v4f*)rqp + 1);
        if (it + 1 < kIters)
            __builtin_prefetch(rqp + kRowsPerIter * D_, 0, 0); // global_prefetch_b8

        // gathered item row (keep temporal: table is L2-resident, 102MB < 192MB L2)
        const int    id  = ids[rowidx];
        const float* itp = itab + (size_t)id * D_ + col;
        const v4f t0 = *(const v4f*)itp;
        const v4f t1 = *(const v4f*)(itp + 4);

        // pos row from LDS stage
        const float* pp = smem + POS_F + wid * D_ + col;
        const v4f p0 = *(const v4f*)pp;
        const v4f p1 = *(const v4f*)(pp + 4);

        float v[8];
        v[0] = r0.x + t0.x + p0.x;  v[1] = r0.y + t0.y + p0.y;
        v[2] = r0.z + t0.z + p0.z;  v[3] = r0.w + t0.w + p0.w;
        v[4] = r1.x + t1.x + p1.x;  v[5] = r1.y + t1.y + p1.y;
        v[6] = r1.z + t1.z + p1.z;  v[7] = r1.w + t1.w + p1.w;

        // ---- segment scatter-add into LDS (ds_add_f32, bank-conflict-free) ----
        const int c = labels[rowidx];
        float* acc = smem + SUM_F + c * D_ + col;
#pragma unroll
        for (int k = 0; k < 8; ++k) atomicAdd(acc + k, v[k]);
        if (lane == 0) atomicAdd(smem + CNT_F + c, 1.0f);

        // ---- LayerNorm: wave32 reduction over the 256-wide row ----
        float s = 0.0f, ss = 0.0f;
#pragma unroll
        for (int k = 0; k < 8; ++k) { s += v[k]; ss += v[k] * v[k]; }
#pragma unroll
        for (int m = 16; m >= 1; m >>= 1) {
            s  += __shfl_xor(s,  m, 32);
            ss += __shfl_xor(ss, m, 32);
        }
        const float mu   = s * (1.0f / D_);
        const float var  = fmaxf(ss * (1.0f / D_) - mu * mu, 0.0f);
        const float rstd = rsqrtf(var + 1e-12f);

        v4f o0, o1;
        o0.x = (v[0] - mu) * rstd * g0.x + e0.x;
        o0.y = (v[1] - mu) * rstd * g0.y + e0.y;
        o0.z = (v[2] - mu) * rstd * g0.z + e0.z;
        o0.w = (v[3] - mu) * rstd * g0.w + e0.w;
        o1.x = (v[4] - mu) * rstd * g1.x + e1.x;
        o1.y = (v[5] - mu) * rstd * g1.y + e1.y;
        o1.z = (v[6] - mu) * rstd * g1.z + e1.z;
        o1.w = (v[7] - mu) * rstd * g1.w + e1.w;

        v4f* op = (v4f*)(out + rowidx * D_ + col);
        __builtin_nontemporal_store(o0, op);       // write-once output: NT b128
        __builtin_nontemporal_store(o1, op + 1);

        __syncthreads();   // all waves done reading pos stage before next overwrite
    }

    // ---- flush cluster means: cluster_emb[b,c,:] = sums[c,:]/max(count,1) ----
    float* cl = out + (size_t)B_ * L_ * D_ + (size_t)b * C_ * D_;
    for (int idx = tid; idx < (C_ * D_) / 4; idx += kBlock) {
        const int   c     = idx >> 6;                 // 64 float4 per cluster row
        const float cnt   = smem[CNT_F + c];
        const float scale = 1.0f / fmaxf(cnt, 1.0f);
        const v4f   sv    = *(const v4f*)(smem + SUM_F + idx * 4);
        const v4f   ov    = sv * scale;
        __builtin_nontemporal_store(ov, (v4f*)cl + idx);
    }
}

extern "C" void kernel_launch(void* const* d_in, const int* in_sizes, int n_in,
                              void* d_out, int out_size, void* d_ws, size_t ws_size,
                              hipStream_t stream) {
    const int*   ids    = (const int*)d_in[0];   // input_ids      [B,L]
    const int*   labels = (const int*)d_in[1];   // cluster_labels [B,L]
    const float* rq     = (const float*)d_in[2]; // rq_item_emb    [B,L,D]
    const float* itab   = (const float*)d_in[3]; // item_table     [V,D]
    const float* ptab   = (const float*)d_in[4]; // pos_table      [L,D]
    const float* gamma  = (const float*)d_in[5]; // ln_gamma       [D]
    const float* beta   = (const float*)d_in[6]; // ln_beta        [D]
    float*       out    = (float*)d_out;         // [B*L*D] seq ++ [B*C*D] cluster

    (void)hipFuncSetAttribute((const void*)s3rec_fused,
                              hipFuncAttributeMaxDynamicSharedMemorySize,
                              (int)SMEM_BYTES);

    hipLaunchKernelGGL(s3rec_fused, dim3(B_), dim3(kBlock), SMEM_BYTES, stream,
                       ids, labels, rq, itab, ptab, gamma, beta, out);
}